// HornerSparseIteration_sparse_23510650978741
// MI455X (gfx1250) — compile-verified
//
#include <hip/hip_runtime.h>
#include <hip/hip_bf16.h>

typedef float v2f __attribute__((ext_vector_type(2)));
typedef float v8f __attribute__((ext_vector_type(8)));
typedef int   v4i __attribute__((__vector_size__(4 * sizeof(int))));

#define CDIV(a,b) (((a)+(b)-1)/(b))

// ------------------- CDNA5 async global->LDS copy helpers -------------------
#if __has_builtin(__builtin_amdgcn_global_load_async_to_lds_b128)
#define HAVE_ASYNC_LDS 1
#else
#define HAVE_ASYNC_LDS 0
#endif

__device__ __forceinline__ void async_copy16(const float* g, float* l) {
#if HAVE_ASYNC_LDS
  // GLOBAL_LOAD_ASYNC_TO_LDS_B128: per-lane 16B global->LDS, tracked by ASYNCcnt
  __builtin_amdgcn_global_load_async_to_lds_b128(
      (__attribute__((address_space(1))) v4i*)(__attribute__((address_space(1))) void*)g,
      (__attribute__((address_space(3))) v4i*)(__attribute__((address_space(3))) void*)l,
      0, 0);
#else
  *(float4*)l = *(const float4*)g;   // global_load_b128 + ds_store_b128
#endif
}

__device__ __forceinline__ void wait_async() {
#if HAVE_ASYNC_LDS
#if __has_builtin(__builtin_amdgcn_s_wait_asynccnt)
  __builtin_amdgcn_s_wait_asynccnt(0);
#else
  asm volatile("s_wait_asynccnt 0x0" ::: "memory");
#endif
#endif
}

// ---------------------------------------------------------------------------
// C[M,N] = A[M,K] @ B[N,K]^T (+bias[n]) using V_WMMA_F32_16X16X4_F32,
// LDS-staged and double-buffered with async global->LDS copies.
// Block = 256 threads = 8 waves. Block tile = 128(M) x 64(N), K-tile = 32.
// Wave w: rows m = w*16, all 64 cols (4 accumulators).
// Fragment layouts (per ISA 7.12.2):
//   A (16x4): lane<16 -> row=lane, K=k0+{0,1}; lane>=16 -> row=lane-16, K=k0+{2,3}
//   B (4x16): lane<16 -> col=lane, K=k0+{0,1}; lane>=16 -> col=lane-16, K=k0+{2,3}
//   C/D: VGPR j, lane<16 -> (m0+j, n); lane>=16 -> (m0+j+8, n)
// Requires M%128==0, N%64==0, K%32==0 (true for all call sites).
// ---------------------------------------------------------------------------
#define GBM 128
#define GBN 64
#define GBK 32
#define GPK 36   // padded LDS row stride (floats): conflict-free, 16B-aligned rows

__global__ __launch_bounds__(256) void gemm_abT_wmma(
    const float* __restrict__ A, const float* __restrict__ B,
    const float* __restrict__ bias, float* __restrict__ C,
    int M, int N, int K)
{
  __shared__ float sA[2][GBM * GPK];
  __shared__ float sB[2][GBN * GPK];

  const int tid  = threadIdx.x;
  const int wave = tid >> 5;
  const int lane = tid & 31;
  const int lr   = lane & 15;
  const int hi   = lane >> 4;

  const int mBlock = blockIdx.y * GBM;
  const int nBlock = blockIdx.x * GBN;

  // cooperative tile load mapping (float4 granularity, fully coalesced)
  const int trow = tid >> 3;          // 0..31
  const int tc   = (tid & 7) * 4;     // float column within K-tile

  auto load_tile = [&](int buf, int kt) {
    const float* Ag = A + (size_t)mBlock * K + kt * GBK;
    const float* Bg = B + (size_t)nBlock * K + kt * GBK;
    #pragma unroll
    for (int i = 0; i < 4; ++i) {           // 128 rows of A
      int r = trow + 32 * i;
      async_copy16(Ag + (size_t)r * K + tc, &sA[buf][r * GPK + tc]);
    }
    #pragma unroll
    for (int i = 0; i < 2; ++i) {           // 64 rows of B
      int r = trow + 32 * i;
      async_copy16(Bg + (size_t)r * K + tc, &sB[buf][r * GPK + tc]);
    }
  };

  v8f acc0 = {}, acc1 = {}, acc2 = {}, acc3 = {};

  load_tile(0, 0);
  wait_async();
  __syncthreads();

  const int KT = K / GBK;
  for (int kt = 0; kt < KT; ++kt) {
    const int cur = kt & 1;
    if (kt + 1 < KT) load_tile(cur ^ 1, kt + 1);   // stream next tile during compute

    const float* la  = &sA[cur][(wave * 16 + lr) * GPK + 2 * hi];
    const float* lb0 = &sB[cur][( 0 + lr) * GPK + 2 * hi];
    const float* lb1 = &sB[cur][(16 + lr) * GPK + 2 * hi];
    const float* lb2 = &sB[cur][(32 + lr) * GPK + 2 * hi];
    const float* lb3 = &sB[cur][(48 + lr) * GPK + 2 * hi];

    #pragma unroll
    for (int kk = 0; kk < GBK; kk += 4) {
      v2f a  = *(const v2f*)(la  + kk);
      v2f b0 = *(const v2f*)(lb0 + kk);
      v2f b1 = *(const v2f*)(lb1 + kk);
      v2f b2 = *(const v2f*)(lb2 + kk);
      v2f b3 = *(const v2f*)(lb3 + kk);
      acc0 = __builtin_amdgcn_wmma_f32_16x16x4_f32(false, a, false, b0, (short)0, acc0, false, false);
      acc1 = __builtin_amdgcn_wmma_f32_16x16x4_f32(false, a, false, b1, (short)0, acc1, false, false);
      acc2 = __builtin_amdgcn_wmma_f32_16x16x4_f32(false, a, false, b2, (short)0, acc2, false, false);
      acc3 = __builtin_amdgcn_wmma_f32_16x16x4_f32(false, a, false, b3, (short)0, acc3, false, false);
    }

    wait_async();       // drain our async copies for the next buffer
    __syncthreads();    // make all waves' LDS writes visible
  }

  const int m0 = mBlock + wave * 16;
  const int cm = m0 + (hi ? 8 : 0);
  v8f accs[4] = {acc0, acc1, acc2, acc3};
  #pragma unroll
  for (int t = 0; t < 4; ++t) {
    int cn = nBlock + t * 16 + lr;
    float bv = bias ? bias[cn] : 0.0f;
    #pragma unroll
    for (int j = 0; j < 8; ++j)
      C[(size_t)(cm + j) * N + cn] = accs[t][j] + bv;
  }
}

// --------------------------- reductions ------------------------------------
__global__ __launch_bounds__(256) void sumsq_kernel(const float* __restrict__ x,
                                                    size_t n, float* out)
{
  __shared__ float red[256];
  float v = 0.f;
  for (size_t j = (size_t)blockIdx.x * 256 + threadIdx.x; j < n;
       j += (size_t)gridDim.x * 256) { float t = x[j]; v += t * t; }
  red[threadIdx.x] = v; __syncthreads();
  for (int s = 128; s > 0; s >>= 1) {
    if (threadIdx.x < s) red[threadIdx.x] += red[threadIdx.x + s];
    __syncthreads();
  }
  if (threadIdx.x == 0) atomicAdd(out, red[0]);
}

__global__ __launch_bounds__(256) void scale_rsqrt_kernel(float* x, const float* s, size_t n)
{
  size_t i = (size_t)blockIdx.x * 256 + threadIdx.x;
  if (i < n) x[i] *= rsqrtf(s[0]);
}

// ks_sum[col] = sum_r kn[r, col]  (one block per column)
__global__ __launch_bounds__(256) void colsum_kernel(const float* __restrict__ kn,
                                                     float* ks_sum, int rows)
{
  int col = blockIdx.x, t = threadIdx.x;
  float s = 0.f;
  for (int r = t; r < rows; r += 256) s += kn[(size_t)r * 256 + col];
  __shared__ float red[256];
  red[t] = s; __syncthreads();
  for (int st = 128; st > 0; st >>= 1) {
    if (t < st) red[t] += red[t + st];
    __syncthreads();
  }
  if (t == 0) ks_sum[col] = red[0];
}

// q'[r,h*64+d] = qn / (normalizer[r,h] * H), qn = q * rsqrt(chunk ssq),
// normalizer = dot(qn[r,h,:], ks_sum[h,:]) + R
__global__ __launch_bounds__(256) void qprime_kernel(const float* __restrict__ q,
    const float* __restrict__ scal, const float* __restrict__ ks_sum,
    float* __restrict__ qp, int Rchunk)
{
  int r = blockIdx.x, t = threadIdx.x;
  float qs = rsqrtf(scal[1 + (r >= Rchunk ? 1 : 0)]);
  float qn = q[(size_t)r * 256 + t] * qs;
  __shared__ float prod[256];
  __shared__ float hnorm[4];
  prod[t] = qn * ks_sum[t];
  __syncthreads();
  if (t < 4) {
    float s = 0.f;
    for (int d = 0; d < 64; ++d) s += prod[t * 64 + d];
    hnorm[t] = (s + 4096.0f) * 4.0f;   // (+R) * H
  }
  __syncthreads();
  qp[(size_t)r * 256 + t] = qn / hnorm[t >> 6];
}

// ---------------- fused gumbel + softmax + top-10 (row resident in regs) ----
__global__ __launch_bounds__(256) void topk_kernel(const float* __restrict__ attn0,
    const float* __restrict__ gum_u, int* __restrict__ cols0,
    float* __restrict__ svals, int Ncols)
{
  int r = blockIdx.x, t = threadIdx.x;
  const float* ar = attn0 + (size_t)r * Ncols;
  const float* gr = gum_u + (size_t)r * Ncols;
  float z[32];
  float lmax = -3.4e38f;
  #pragma unroll
  for (int j = 0; j < 32; ++j) {
    int c = t + j * 256;
    float zz = ar[c] - __logf(-__logf(gr[c]));
    z[j] = zz; lmax = fmaxf(lmax, zz);
  }
  __shared__ float red[256];
  __shared__ float bval[256];
  __shared__ int   bidx[256];
  __shared__ int   chosen[10];
  red[t] = lmax; __syncthreads();
  for (int s = 128; s > 0; s >>= 1) { if (t < s) red[t] = fmaxf(red[t], red[t + s]); __syncthreads(); }
  float m = red[0]; __syncthreads();
  float lsum = 0.f;
  #pragma unroll
  for (int j = 0; j < 32; ++j) lsum += __expf(z[j] - m);
  red[t] = lsum; __syncthreads();
  for (int s = 128; s > 0; s >>= 1) { if (t < s) red[t] += red[t + s]; __syncthreads(); }
  float ssum = red[0]; __syncthreads();

  for (int sel = 0; sel < 10; ++sel) {
    float best = -3.4e38f; int bi = 0x7fffffff;
    #pragma unroll
    for (int j = 0; j < 32; ++j) {
      int c = t + j * 256;
      bool taken = false;
      for (int s2 = 0; s2 < sel; ++s2) taken |= (chosen[s2] == c);
      if (!taken && z[j] > best) { best = z[j]; bi = c; }
    }
    bval[t] = best; bidx[t] = bi; __syncthreads();
    for (int s = 128; s > 0; s >>= 1) {
      if (t < s) {
        if (bval[t + s] > bval[t] || (bval[t + s] == bval[t] && bidx[t + s] < bidx[t])) {
          bval[t] = bval[t + s]; bidx[t] = bidx[t + s];
        }
      }
      __syncthreads();
    }
    if (t == 0) {
      chosen[sel] = bidx[0];
      cols0[r * 10 + sel] = bidx[0];
      svals[(size_t)r * 10 + sel] = __expf(bval[0] - m) / ssum;
    }
    __syncthreads();
  }
}

// chunk-1 values only at chunk-0's top-k columns: one wave per (r,t) dot product
__global__ __launch_bounds__(256) void rest_kernel(const float* __restrict__ qp,
    const float* __restrict__ kn, const int* __restrict__ cols0,
    float* __restrict__ svals, int Rr, int K)
{
  int gw = (int)((blockIdx.x * 256 + threadIdx.x) >> 5);
  int lane = threadIdx.x & 31;
  if (gw >= Rr * 10) return;
  int r = gw / 10, t = gw % 10;
  const float* a = qp + (size_t)(Rr + r) * K;
  const float* b = kn + (size_t)cols0[r * 10 + t] * K;
  float s = 0.f;
  for (int c = lane; c < K; c += 32) s += a[c] * b[c];
  for (int off = 16; off > 0; off >>= 1) s += __shfl_xor(s, off, 32);
  if (lane == 0) svals[(size_t)(Rr + r) * 10 + t] = s;
}

// ------------------------------ sparse series -------------------------------
__global__ __launch_bounds__(256) void spmm_ell_kernel(const float* __restrict__ svals,
    const int* __restrict__ cols0, const float* __restrict__ x,
    float* __restrict__ y, int Nn, int Rr)
{
  int i = blockIdx.x * 256 + threadIdx.x;
  if (i >= Nn * 16) return;
  int r = i >> 4, c = i & 15;
  const int*   cc = cols0 + (size_t)(r % Rr) * 10;
  const float* vv = svals + (size_t)r * 10;
  float acc = 0.f;
  #pragma unroll
  for (int t = 0; t < 10; ++t) acc += vv[t] * x[(size_t)cc[t] * 16 + c];
  y[i] = acc;
}

__global__ __launch_bounds__(256) void spmm_coo_kernel(const int* __restrict__ rows,
    const int* __restrict__ cols, const float* __restrict__ vals,
    const float* __restrict__ x, float* __restrict__ y, int nnz)
{
  int e = blockIdx.x * 256 + threadIdx.x;
  if (e >= nnz) return;
  int rr = rows[e], cc = cols[e];
  float v = vals[e];
  const float* xc = x + (size_t)cc * 16;
  float* yr = y + (size_t)rr * 16;
  #pragma unroll
  for (int c = 0; c < 16; ++c) atomicAdd(&yr[c], v * xc[c]);
}

__global__ __launch_bounds__(256) void scale_copy_kernel(float* dst, const float* src,
                                                         const float* w, int n)
{
  int i = blockIdx.x * 256 + threadIdx.x;
  if (i < n) dst[i] = w[0] * src[i];
}

__global__ __launch_bounds__(256) void axpy_kernel(float* acc, const float* x,
                                                   const float* w, int n)
{
  int i = blockIdx.x * 256 + threadIdx.x;
  if (i < n) acc[i] += w[0] * x[i];
}

__global__ __launch_bounds__(256) void gather_kernel(const float* __restrict__ acc,
    const int* __restrict__ idx, float* __restrict__ out, int n)
{
  int i = blockIdx.x * 256 + threadIdx.x;
  if (i >= n) return;
  out[i] = acc[(size_t)idx[i >> 4] * 16 + (i & 15)];
}

// ---------------------------------------------------------------------------
extern "C" void kernel_launch(void* const* d_in, const int* in_sizes, int n_in,
                              void* d_out, int out_size, void* d_ws, size_t ws_size,
                              hipStream_t stream)
{
  const float* local_preds = (const float*)d_in[0];
  const float* fea   = (const float*)d_in[1];
  const float* Wq_w  = (const float*)d_in[2];
  const float* Wq_b  = (const float*)d_in[3];
  const float* Wk_w  = (const float*)d_in[4];
  const float* Wk_b  = (const float*)d_in[5];
  const float* lin1  = (const float*)d_in[6];
  const float* lin2  = (const float*)d_in[7];
  const float* gum_u = (const float*)d_in[8];
  const float* a_vals = (const float*)d_in[9];
  const int*   a_rows = (const int*)d_in[10];
  const int*   a_cols = (const int*)d_in[11];
  const int*   idx    = (const int*)d_in[12];
  int nnz = in_sizes[9];

  constexpr int Nn = 8192, Fk = 256, Rr = 4096, NC = 16;
  const int nElem = Nn * NC;

  float* ws = (float*)d_ws;
  size_t o = 0;
  float* q      = ws + o; o += (size_t)Nn * Fk;      // 2M
  float* kn     = ws + o; o += (size_t)Nn * Fk;      // 2M (k, then normalized in place)
  float* qp     = ws + o; o += (size_t)Nn * Fk;      // 2M
  float* scal   = ws + o; o += 16;                   // [0]=||k||^2, [1]=q ssq c0, [2]=c1
  float* ks_sum = ws + o; o += 256;
  float* attn0  = ws + o; o += (size_t)Rr * Nn;      // 33.5M floats = 128 MB
  int*   cols0  = (int*)(ws + o); o += (size_t)Rr * 10;
  float* svals  = ws + o; o += (size_t)Nn * 10;
  float* bufA   = ws + o; o += nElem;                // accumulator
  float* bufB   = ws + o; o += nElem;                // tmp
  float* bufC   = ws + o; o += nElem;                // scratch / ping-pong

  dim3 blk(256);

  // --- Q/K projections: C = fea @ W^T + b  (WMMA f32 16x16x4, async-LDS) ---
  gemm_abT_wmma<<<dim3(Fk / 64, Nn / 128), blk, 0, stream>>>(fea, Wq_w, Wq_b, q,  Nn, Fk, Fk);
  gemm_abT_wmma<<<dim3(Fk / 64, Nn / 128), blk, 0, stream>>>(fea, Wk_w, Wk_b, kn, Nn, Fk, Fk);

  // --- norms ---
  (void)hipMemsetAsync(scal, 0, 3 * sizeof(float), stream);
  sumsq_kernel<<<512, blk, 0, stream>>>(kn, (size_t)Nn * Fk, scal + 0);
  sumsq_kernel<<<256, blk, 0, stream>>>(q,                    (size_t)Rr * Fk, scal + 1);
  sumsq_kernel<<<256, blk, 0, stream>>>(q + (size_t)Rr * Fk,  (size_t)Rr * Fk, scal + 2);
  scale_rsqrt_kernel<<<CDIV((size_t)Nn * Fk, 256), blk, 0, stream>>>(kn, scal, (size_t)Nn * Fk);
  colsum_kernel<<<Fk, blk, 0, stream>>>(kn, ks_sum, Nn);
  qprime_kernel<<<Nn, blk, 0, stream>>>(q, scal, ks_sum, qp, Rr);

  // --- chunk-0 attention GEMM (chunk 1 never materialized) ---
  gemm_abT_wmma<<<dim3(Nn / 64, Rr / 128), blk, 0, stream>>>(qp, kn, nullptr, attn0, Rr, Nn, Fk);

  // --- gumbel softmax + top-10, then chunk-1 gathered dot products ---
  topk_kernel<<<Rr, blk, 0, stream>>>(attn0, gum_u, cols0, svals, Nn);
  rest_kernel<<<CDIV(Rr * 10 * 32, 256), blk, 0, stream>>>(qp, kn, cols0, svals, Rr, Fk);

  // --- power series 1: sparse attention applied twice per step ---
  (void)hipMemcpyAsync(bufB, local_preds, nElem * sizeof(float), hipMemcpyDeviceToDevice, stream);
  scale_copy_kernel<<<CDIV(nElem, 256), blk, 0, stream>>>(bufA, local_preds, lin2 + 0, nElem);
  for (int i = 1; i < 8; ++i) {
    spmm_ell_kernel<<<CDIV(nElem, 256), blk, 0, stream>>>(svals, cols0, bufB, bufC, Nn, Rr);
    spmm_ell_kernel<<<CDIV(nElem, 256), blk, 0, stream>>>(svals, cols0, bufC, bufB, Nn, Rr);
    axpy_kernel<<<CDIV(nElem, 256), blk, 0, stream>>>(bufA, bufB, lin2 + i, nElem);
  }

  // --- power series 2: normalized adjacency (COO, atomic scatter) ---
  (void)hipMemcpyAsync(bufB, bufA, nElem * sizeof(float), hipMemcpyDeviceToDevice, stream);
  scale_copy_kernel<<<CDIV(nElem, 256), blk, 0, stream>>>(bufA, bufA, lin1 + 0, nElem);
  float* tin = bufB; float* tout = bufC;
  for (int i = 1; i < 8; ++i) {
    (void)hipMemsetAsync(tout, 0, nElem * sizeof(float), stream);
    spmm_coo_kernel<<<CDIV(nnz, 256), blk, 0, stream>>>(a_rows, a_cols, a_vals, tin, tout, nnz);
    axpy_kernel<<<CDIV(nElem, 256), blk, 0, stream>>>(bufA, tout, lin1 + i, nElem);
    float* sw = tin; tin = tout; tout = sw;
  }

  // --- final gather preds[idx] ---
  gather_kernel<<<CDIV(out_size, 256), blk, 0, stream>>>(bufA, idx, (float*)d_out, out_size);
}